// SAGE_87591563034885
// MI455X (gfx1250) — compile-verified
//
#include <hip/hip_runtime.h>

// ---------------------------------------------------------------------------
// Residual GraphSAGE, 3 layers, N=100000 nodes, D=128, E=1,600,000 edges.
//
// MI455X reasoning:
//  * Node features = 100000*128*4 = 51.2 MB -> fits entirely in 192 MB L2.
//    The edge gather + scatter-add (204.8M f32 atomics/layer) is L2-resident;
//    we use native GLOBAL_ATOMIC_ADD_F32 (no-return, fire-and-forget) via
//    unsafeAtomicAdd, one wave32 per edge, float4 per lane (512B coalesced
//    per wave).
//  * GEMMs (mean@Wl + x@Wr, 6.6 GFLOP/layer) are tiny vs the scatter, so we
//    keep full f32 precision using V_WMMA_F32_16X16X4_F32 (wave32 matrix op)
//    instead of risking f16 truncation error for no real speedup.
// ---------------------------------------------------------------------------

typedef __attribute__((ext_vector_type(2))) float v2f;
typedef __attribute__((ext_vector_type(8))) float v8f;

#define N_NODES 100000
#define N_EDGES 1600000
#define D       128
#define N_LAYERS 3

// ---- zero agg[N*D] and cnt[N] ---------------------------------------------
__global__ __launch_bounds__(256) void sage_zero_kernel(float* __restrict__ agg,
                                                        float* __restrict__ cnt) {
    size_t i = (size_t)blockIdx.x * blockDim.x + threadIdx.x;
    if (i < (size_t)N_NODES * D) agg[i] = 0.0f;
    if (i < (size_t)N_NODES)     cnt[i] = 0.0f;
}

// ---- edge scatter: one wave32 per edge, float4 per lane -------------------
__global__ __launch_bounds__(256) void sage_scatter_kernel(
        const float* __restrict__ h,
        const int*   __restrict__ src,
        const int*   __restrict__ dst,
        float*       __restrict__ agg,
        float*       __restrict__ cnt) {
    int wave = (int)(((size_t)blockIdx.x * blockDim.x + threadIdx.x) >> 5);
    int lane = threadIdx.x & 31;
    if (wave >= N_EDGES) return;                 // wave-uniform predicate

    int s = src[wave];
    int d = dst[wave];

    // gather 512B of the source row per wave (L2-resident)
    const float4* hp = (const float4*)(h + (size_t)s * D);
    float4 v = hp[lane];

    // native f32 atomic adds into L2 (GLOBAL_ATOMIC_ADD_F32, no CAS loop)
    float* ap = agg + (size_t)d * D + (size_t)lane * 4;
    unsafeAtomicAdd(ap + 0, v.x);
    unsafeAtomicAdd(ap + 1, v.y);
    unsafeAtomicAdd(ap + 2, v.z);
    unsafeAtomicAdd(ap + 3, v.w);

    if (lane == 0)
        unsafeAtomicAdd(cnt + d, 1.0f);
}

// ---- agg -> mean (in place) -----------------------------------------------
__global__ __launch_bounds__(256) void sage_mean_kernel(float* __restrict__ agg,
                                                        const float* __restrict__ cnt) {
    size_t i = (size_t)blockIdx.x * blockDim.x + threadIdx.x;
    if (i >= (size_t)N_NODES * D) return;
    float c = cnt[i >> 7];                       // D == 128
    agg[i] = agg[i] / fmaxf(c, 1.0f);
}

// ---- fused dual-GEMM + bias + residual (+ReLU) via V_WMMA_F32_16X16X4_F32 -
// One wave computes one 16x16 output tile:  D = mean@Wl + x@Wr  (accumulated
// in a single v8f), then epilogue adds bl + x residual.
//
// f32 WMMA operand layout (wave32):
//   A (16x4):  lanes 0-15 -> M=lane,   VGPR0/1 = K=0/1
//              lanes16-31 -> M=lane-16,VGPR0/1 = K=2/3
//   B (4x16):  lanes 0-15 -> N=lane,   VGPR0/1 = K=0/1
//              lanes16-31 -> N=lane-16,VGPR0/1 = K=2/3
//   C/D(16x16):VGPR r -> M=r (lanes 0-15) / M=r+8 (lanes 16-31), N=lane&15
__global__ __launch_bounds__(256) void sage_gemm_kernel(
        const float* __restrict__ mean,
        const float* __restrict__ xin,
        const float* __restrict__ Wl,
        const float* __restrict__ bl,
        const float* __restrict__ Wr,
        float*       __restrict__ hout,
        int do_relu) {
    const int NT = D / 16;                       // 8 column tiles
    int waveId = (int)(((size_t)blockIdx.x * blockDim.x + threadIdx.x) >> 5);
    int mt = waveId / NT;
    int nt = waveId % NT;
    if (mt >= N_NODES / 16) return;              // wave-uniform (EXEC stays full)

    int lane = threadIdx.x & 31;
    int half = lane >> 4;                        // 0: K=0/1, 1: K=2/3
    int l15  = lane & 15;
    int row0 = mt * 16;
    int n0   = nt * 16;

    const float* mrow = mean + (size_t)(row0 + l15) * D;
    const float* xrow = xin  + (size_t)(row0 + l15) * D;

    v8f acc = {};
#pragma unroll 4
    for (int k0 = 0; k0 < D; k0 += 4) {
        int ka = k0 + 2 * half;
        v2f aM = *(const v2f*)(mrow + ka);       // A tile of mean
        v2f aX = *(const v2f*)(xrow + ka);       // A tile of x
        v2f bL, bR;
        bL.x = Wl[(size_t)(ka + 0) * D + n0 + l15];
        bL.y = Wl[(size_t)(ka + 1) * D + n0 + l15];
        bR.x = Wr[(size_t)(ka + 0) * D + n0 + l15];
        bR.y = Wr[(size_t)(ka + 1) * D + n0 + l15];
        // acc += aM * bL ; acc += aX * bR   (full f32 tensor path)
        acc = __builtin_amdgcn_wmma_f32_16x16x4_f32(
                false, aM, false, bL, (short)0, acc, false, false);
        acc = __builtin_amdgcn_wmma_f32_16x16x4_f32(
                false, aX, false, bR, (short)0, acc, false, false);
    }

    // epilogue: + bl + x residual (+ ReLU), write 8 elements per lane
    int   col  = n0 + l15;
    float bias = bl[col];
#pragma unroll
    for (int r = 0; r < 8; ++r) {
        int   orow = row0 + r + 8 * half;
        float val  = acc[r] + bias + xin[(size_t)orow * D + col];
        if (do_relu) val = fmaxf(val, 0.0f);
        hout[(size_t)orow * D + col] = val;
    }
}

// ---------------------------------------------------------------------------
extern "C" void kernel_launch(void* const* d_in, const int* in_sizes, int n_in,
                              void* d_out, int out_size, void* d_ws, size_t ws_size,
                              hipStream_t stream) {
    const float* x    = (const float*)d_in[0];          // [N, D]
    const int*   eidx = (const int*)  d_in[1];          // [2, E] row-major
    const float* Wl   = (const float*)d_in[2];          // [3, D, D]
    const float* bl   = (const float*)d_in[3];          // [3, D]
    const float* Wr   = (const float*)d_in[4];          // [3, D, D]
    float* out = (float*)d_out;                         // [N, D]

    const int* src = eidx;                              // edge_index[0]
    const int* dst = eidx + N_EDGES;                    // edge_index[1]

    // workspace layout: agg[N*D] | cnt[N] | hA[N*D] | hB[N*D]
    float* agg = (float*)d_ws;
    float* cnt = agg + (size_t)N_NODES * D;
    float* hA  = cnt + N_NODES;
    float* hB  = hA + (size_t)N_NODES * D;

    const int zeroBlocks    = (int)(((size_t)N_NODES * D + 255) / 256);
    const int scatterBlocks = (int)(((size_t)N_EDGES * 32 + 255) / 256);
    const int meanBlocks    = zeroBlocks;
    const int tiles         = (N_NODES / 16) * (D / 16);          // 50000 waves
    const int gemmBlocks    = (tiles + 7) / 8;                    // 8 waves/block

    const float* hin = x;
    for (int layer = 0; layer < N_LAYERS; ++layer) {
        float* hout = (layer == 0) ? hA : (layer == 1) ? hB : out;

        sage_zero_kernel<<<zeroBlocks, 256, 0, stream>>>(agg, cnt);
        sage_scatter_kernel<<<scatterBlocks, 256, 0, stream>>>(hin, src, dst, agg, cnt);
        sage_mean_kernel<<<meanBlocks, 256, 0, stream>>>(agg, cnt);
        sage_gemm_kernel<<<gemmBlocks, 256, 0, stream>>>(
                agg, hin,
                Wl + (size_t)layer * D * D,
                bl + (size_t)layer * D,
                Wr + (size_t)layer * D * D,
                hout,
                (layer != N_LAYERS - 1) ? 1 : 0);

        hin = hout;
    }
}